// AFTFull_Parameterized_214748365538
// MI455X (gfx1250) — compile-verified
//
#include <hip/hip_runtime.h>

// AFT-Full, factorized: exp(k + w) = exp(w) * exp(k)  =>  num/den become GEMMs.
// All GEMMs use V_WMMA_F32_16X16X4_F32 (exact fp32 matrix op on CDNA5).

#define B_    2
#define TQ_   512
#define TKV_  512
#define DIM_  512
#define HID_  128
#define BDIM_ 128

typedef __attribute__((ext_vector_type(2))) float v2f;
typedef __attribute__((ext_vector_type(8))) float v8f;

#define WMMA_F32(a, b, c) \
  __builtin_amdgcn_wmma_f32_16x16x4_f32(false, (a), false, (b), (short)0, (c), false, false)

// ---------------------------------------------------------------------------
// Kernel 1: Wexp[t,s] = exp( (u @ v)[t,s] )   u:(TQ,BDIM) row-major, v:(BDIM,TKV)
// ---------------------------------------------------------------------------
__global__ __launch_bounds__(128) void aft_bias_kernel(
    const float* __restrict__ u, const float* __restrict__ vm,
    float* __restrict__ Wexp) {
  const int wave = blockIdx.x * (blockDim.x >> 5) + (threadIdx.x >> 5);
  const int l = threadIdx.x & 31;
  const int m = l & 15;          // M row / N col within tile
  const int half = l >> 4;       // K half select
  const int tiles_n = TKV_ / 16; // 32
  const int tm = wave / tiles_n;
  const int tn = wave % tiles_n;
  const int m0 = tm * 16, n0 = tn * 16;

  v8f acc = {};
  for (int k0 = 0; k0 < BDIM_; k0 += 4) {
    const int ka = k0 + 2 * half;
    v2f a, b;
    a.x = u[(m0 + m) * BDIM_ + ka];
    a.y = u[(m0 + m) * BDIM_ + ka + 1];
    b.x = vm[ka * TKV_ + (n0 + m)];
    b.y = vm[(ka + 1) * TKV_ + (n0 + m)];
    acc = WMMA_F32(a, b, acc);
  }
#pragma unroll
  for (int r = 0; r < 8; ++r) {
    const int row = m0 + r + 8 * half;
    Wexp[row * TKV_ + n0 + m] = __expf(acc[r]);
  }
}

// ---------------------------------------------------------------------------
// Kernel 2: S[b,t,h] = sigmoid( (q @ qW^T)[b,t,h] + qb[h] )
//   q:(B,TQ,DIM), qW:(HID,DIM) row-major -> B-frag element (k,n) = qW[n*DIM+k]
// ---------------------------------------------------------------------------
__global__ __launch_bounds__(128) void aft_qproj_kernel(
    const float* __restrict__ q, const float* __restrict__ qW,
    const float* __restrict__ qb, float* __restrict__ S) {
  const int wave = blockIdx.x * (blockDim.x >> 5) + (threadIdx.x >> 5);
  const int l = threadIdx.x & 31;
  const int m = l & 15, half = l >> 4;
  const int tiles_n = HID_ / 16;               // 8
  const int tiles_m = TQ_ / 16;                // 32
  const int b  = wave / (tiles_m * tiles_n);
  const int rm = wave % (tiles_m * tiles_n);
  const int m0 = (rm / tiles_n) * 16, n0 = (rm % tiles_n) * 16;
  const float* A = q + (size_t)b * TQ_ * DIM_;

  v8f acc = {};
  for (int k0 = 0; k0 < DIM_; k0 += 4) {
    const int ka = k0 + 2 * half;
    v2f a, bw;
    a.x = A[(m0 + m) * DIM_ + ka];
    a.y = A[(m0 + m) * DIM_ + ka + 1];
    const float* wp = qW + (size_t)(n0 + m) * DIM_ + ka;
    bw.x = wp[0];
    bw.y = wp[1];
    acc = WMMA_F32(a, bw, acc);
  }
  const float bias = qb[n0 + m];
#pragma unroll
  for (int r = 0; r < 8; ++r) {
    const int row = m0 + r + 8 * half;
    const float x = acc[r] + bias;
    S[((size_t)b * TQ_ + row) * HID_ + n0 + m] = 1.0f / (1.0f + __expf(-x));
  }
}

// ---------------------------------------------------------------------------
// Kernel 3: kc = kv@kW^T + kb ; vc = kv@vW^T + vb ; E = exp(kc); EV = E*vc
//   Dual accumulators share the A (kv) fragment -> 2 v_wmma per tile-K step.
// ---------------------------------------------------------------------------
__global__ __launch_bounds__(128) void aft_kvproj_kernel(
    const float* __restrict__ kv,
    const float* __restrict__ kW, const float* __restrict__ kb,
    const float* __restrict__ vW, const float* __restrict__ vb,
    float* __restrict__ E, float* __restrict__ EV) {
  const int wave = blockIdx.x * (blockDim.x >> 5) + (threadIdx.x >> 5);
  const int l = threadIdx.x & 31;
  const int m = l & 15, half = l >> 4;
  const int tiles_n = HID_ / 16;               // 8
  const int tiles_m = TKV_ / 16;               // 32
  const int b  = wave / (tiles_m * tiles_n);
  const int rm = wave % (tiles_m * tiles_n);
  const int m0 = (rm / tiles_n) * 16, n0 = (rm % tiles_n) * 16;
  const float* A = kv + (size_t)b * TKV_ * DIM_;

  v8f acck = {}, accv = {};
  for (int k0 = 0; k0 < DIM_; k0 += 4) {
    const int ka = k0 + 2 * half;
    v2f a, bk_, bv_;
    a.x = A[(m0 + m) * DIM_ + ka];
    a.y = A[(m0 + m) * DIM_ + ka + 1];
    const float* kp = kW + (size_t)(n0 + m) * DIM_ + ka;
    const float* vp = vW + (size_t)(n0 + m) * DIM_ + ka;
    bk_.x = kp[0]; bk_.y = kp[1];
    bv_.x = vp[0]; bv_.y = vp[1];
    acck = WMMA_F32(a, bk_, acck);
    accv = WMMA_F32(a, bv_, accv);
  }
  const float biask = kb[n0 + m];
  const float biasv = vb[n0 + m];
#pragma unroll
  for (int r = 0; r < 8; ++r) {
    const int row = m0 + r + 8 * half;
    const size_t idx = ((size_t)b * TKV_ + row) * HID_ + n0 + m;
    const float e = __expf(acck[r] + biask);
    E[idx]  = e;
    EV[idx] = e * (accv[r] + biasv);
  }
}

// ---------------------------------------------------------------------------
// Kernel 4: out[b,t,h] = S[b,t,h] * (Wexp @ EV)[b,t,h] / (Wexp @ E)[b,t,h]
//   Dual accumulators share the Wexp A fragment.
// ---------------------------------------------------------------------------
__global__ __launch_bounds__(128) void aft_out_kernel(
    const float* __restrict__ Wexp, const float* __restrict__ E,
    const float* __restrict__ EV, const float* __restrict__ S,
    float* __restrict__ out) {
  const int wave = blockIdx.x * (blockDim.x >> 5) + (threadIdx.x >> 5);
  const int l = threadIdx.x & 31;
  const int m = l & 15, half = l >> 4;
  const int tiles_n = HID_ / 16;               // 8
  const int tiles_m = TQ_ / 16;                // 32
  const int b  = wave / (tiles_m * tiles_n);
  const int rm = wave % (tiles_m * tiles_n);
  const int m0 = (rm / tiles_n) * 16, n0 = (rm % tiles_n) * 16;
  const float* Eb  = E  + (size_t)b * TKV_ * HID_;
  const float* EVb = EV + (size_t)b * TKV_ * HID_;

  v8f accn = {}, accd = {};
  for (int k0 = 0; k0 < TKV_; k0 += 4) {
    const int ka = k0 + 2 * half;
    v2f a, b1, b2;
    a.x = Wexp[(m0 + m) * TKV_ + ka];
    a.y = Wexp[(m0 + m) * TKV_ + ka + 1];
    b1.x = EVb[ka * HID_ + n0 + m];
    b1.y = EVb[(ka + 1) * HID_ + n0 + m];
    b2.x = Eb[ka * HID_ + n0 + m];
    b2.y = Eb[(ka + 1) * HID_ + n0 + m];
    accn = WMMA_F32(a, b1, accn);
    accd = WMMA_F32(a, b2, accd);
  }
#pragma unroll
  for (int r = 0; r < 8; ++r) {
    const int row = m0 + r + 8 * half;
    const size_t idx = ((size_t)b * TQ_ + row) * HID_ + n0 + m;
    out[idx] = S[idx] * (accn[r] / accd[r]);
  }
}

// ---------------------------------------------------------------------------
extern "C" void kernel_launch(void* const* d_in, const int* in_sizes, int n_in,
                              void* d_out, int out_size, void* d_ws, size_t ws_size,
                              hipStream_t stream) {
  const float* q     = (const float*)d_in[0];
  const float* kv    = (const float*)d_in[1];
  const float* qW_w  = (const float*)d_in[2];
  const float* qW_b  = (const float*)d_in[3];
  const float* kW_w  = (const float*)d_in[4];
  const float* kW_b  = (const float*)d_in[5];
  const float* vW_w  = (const float*)d_in[6];
  const float* vW_b  = (const float*)d_in[7];
  const float* wbu   = (const float*)d_in[8];
  const float* wbv   = (const float*)d_in[9];
  float* out = (float*)d_out;

  // Workspace layout (floats): Wexp(TQ*TKV) | S(B*TQ*HID) | E(B*TKV*HID) | EV(B*TKV*HID)
  float* Wexp = (float*)d_ws;
  float* S    = Wexp + (size_t)TQ_ * TKV_;
  float* E    = S    + (size_t)B_ * TQ_ * HID_;
  float* EV   = E    + (size_t)B_ * TKV_ * HID_;

  const int threads = 128;              // 4 waves per block
  const int wpb = threads / 32;

  // 1) Wexp: (TQ/16)*(TKV/16) = 1024 waves
  {
    const int waves = (TQ_ / 16) * (TKV_ / 16);
    aft_bias_kernel<<<waves / wpb, threads, 0, stream>>>(wbu, wbv, Wexp);
  }
  // 2) S: B*(TQ/16)*(HID/16) = 512 waves
  {
    const int waves = B_ * (TQ_ / 16) * (HID_ / 16);
    aft_qproj_kernel<<<waves / wpb, threads, 0, stream>>>(q, qW_w, qW_b, S);
  }
  // 3) E, EV: 512 waves
  {
    const int waves = B_ * (TKV_ / 16) * (HID_ / 16);
    aft_kvproj_kernel<<<waves / wpb, threads, 0, stream>>>(kv, kW_w, kW_b, vW_w, vW_b, E, EV);
  }
  // 4) out: 512 waves
  {
    const int waves = B_ * (TQ_ / 16) * (HID_ / 16);
    aft_out_kernel<<<waves / wpb, threads, 0, stream>>>(Wexp, E, EV, S, out);
  }
}